// HaloAttention_79809082294701
// MI455X (gfx1250) — compile-verified
//
#include <hip/hip_runtime.h>
#include <hip/hip_bf16.h>

typedef __attribute__((ext_vector_type(2))) float v2f;
typedef __attribute__((ext_vector_type(8))) float v8f;

static __device__ __forceinline__ v8f wmma4(v2f a, v2f b, v8f c) {
    // D = A(16x4 f32) * B(4x16 f32) + C(16x16 f32); exact f32 on the matrix pipe
    return __builtin_amdgcn_wmma_f32_16x16x4_f32(
        /*neg_a=*/false, a, /*neg_b=*/false, b,
        /*c_mod=*/(short)0, c, /*reuse_a=*/false, /*reuse_b=*/false);
}

#define NUM_HEADS 8
#define BLOCK 8
#define HALO 2
#define KD 32
#define KVK 12
#define NPOS 144          // 12*12
#define HDIM 128
#define CC 256
#define QKVN 768          // 256 (q) + 512 (kv)
#define MROWS 65536       // 4*128*128

// ---------------------------------------------------------------------------
// Weight prep: wcatT[n][k] = concat(w_q, w_kv)[k][n]; woutT[n][k] = w_out[k][n]
// ---------------------------------------------------------------------------
__global__ __launch_bounds__(256) void prep_weights(
    const float* __restrict__ w_q, const float* __restrict__ w_kv,
    const float* __restrict__ w_out,
    float* __restrict__ wcatT, float* __restrict__ woutT)
{
    int i = blockIdx.x * 256 + threadIdx.x;
    if (i < QKVN * CC) {
        int n = i >> 8, k = i & 255;
        wcatT[i] = (n < CC) ? w_q[k * CC + n] : w_kv[k * 512 + (n - CC)];
    }
    if (i < CC * CC) {
        int n = i >> 8, k = i & 255;
        woutT[i] = w_out[k * CC + n];
    }
}

// ---------------------------------------------------------------------------
// GEMM: C[M][Nt] = A[M][256] * BT[Nt][256]^T, f32 WMMA 16x16x4
// block tile 128M x 64N, 8 waves stacked in M, K staged in 64-wide LDS chunks
// ---------------------------------------------------------------------------
__global__ __launch_bounds__(256) void gemm_wmma_f32(
    const float* __restrict__ A, const float* __restrict__ BT,
    float* __restrict__ C, int Nt)
{
    __shared__ float Alds[128 * 64];
    __shared__ float Blds[64 * 64];
    const int tid = threadIdx.x;
    const int lane = tid & 31, wave = tid >> 5;
    const int lmod = lane & 15, lhalf = lane >> 4;
    const long Mbase = (long)blockIdx.x * 128;
    const int Nbase = blockIdx.y * 64;

    v8f acc[4] = {v8f{}, v8f{}, v8f{}, v8f{}};

    for (int kc = 0; kc < 4; ++kc) {
        // stage A chunk: 128 x 64 f32 (2048 float4, coalesced 256B/row)
        #pragma unroll
        for (int i = 0; i < 8; ++i) {
            int f = tid + i * 256;
            int row = f >> 4, c4 = f & 15;
            float4 v = *(const float4*)(A + (Mbase + row) * 256 + kc * 64 + c4 * 4);
            *(float4*)(Alds + row * 64 + c4 * 4) = v;
        }
        // stage B chunk: 64 x 64 f32
        #pragma unroll
        for (int i = 0; i < 4; ++i) {
            int f = tid + i * 256;
            int row = f >> 4, c4 = f & 15;
            float4 v = *(const float4*)(BT + (long)(Nbase + row) * 256 + kc * 64 + c4 * 4);
            *(float4*)(Blds + row * 64 + c4 * 4) = v;
        }
        __syncthreads();
        // A lane layout: M = lane%16, K pair = 2*(lane/16) + 4*step  (ISA 7.12.2)
        #pragma unroll
        for (int step = 0; step < 16; ++step) {
            int k0 = 2 * lhalf + 4 * step;
            v2f a = *(const v2f*)(Alds + (wave * 16 + lmod) * 64 + k0);
            #pragma unroll
            for (int ns = 0; ns < 4; ++ns) {
                v2f b = *(const v2f*)(Blds + (ns * 16 + lmod) * 64 + k0);
                acc[ns] = wmma4(a, b, acc[ns]);
            }
        }
        __syncthreads();
    }
    // C/D layout: VGPR g -> M = g + 8*(lane/16), N = lane%16
    #pragma unroll
    for (int ns = 0; ns < 4; ++ns) {
        #pragma unroll
        for (int g = 0; g < 8; ++g) {
            long row = Mbase + wave * 16 + g + 8 * lhalf;
            int col = Nbase + ns * 16 + lmod;
            C[row * (long)Nt + col] = acc[ns][g];
        }
    }
}

// ---------------------------------------------------------------------------
// Halo attention: one workgroup per (b, head, block_y, block_x)
// ---------------------------------------------------------------------------
__global__ __launch_bounds__(256) void halo_attn(
    const float* __restrict__ qkv,      // [65536][768]
    const float* __restrict__ emb_h,    // [32][23]
    const float* __restrict__ emb_w,    // [32][23]
    float* __restrict__ attnout)        // [65536][256]
{
    // manual LDS layout (floats); score buffer overlaps dead emb staging
    __shared__ float smem[24576];                 // 96 KB
    float* q_lds = smem;                          // [64][32]
    float* k_lds = smem + 2048;                   // [144][32]
    float* vT_lds = smem + 6656;                  // [32][144]
    float* relh  = smem + 11264;                  // [64][32] (r<23 valid)
    float* relw  = smem + 13312;                  // [64][32]
    float* ehT   = smem + 15360;                  // [32][32] emb_h^T, zero pad
    float* ewT   = smem + 16384;                  // [32][32]
    float* sc    = smem + 15360;                  // [64][144] (after rel done)

    const int tid = threadIdx.x;
    const int lane = tid & 31, wave = tid >> 5;
    const int lmod = lane & 15, lhalf = lane >> 4;
    int t = blockIdx.x;
    const int wx = t & 15;  t >>= 4;
    const int hy = t & 15;  t >>= 4;
    const int head = t & 7; t >>= 3;
    const int b = t;

    // ---- load q tile (unscaled; scale folded into scores later)
    #pragma unroll
    for (int i = 0; i < 2; ++i) {
        int f = tid + i * 256;           // float4 index, 8 per query row
        int qrow = f >> 3, c4 = f & 7;
        int qy = qrow >> 3, qx = qrow & 7;
        long grow = ((long)(b * HDIM + hy * 8 + qy)) * HDIM + wx * 8 + qx;
        float4 v = *(const float4*)(qkv + grow * QKVN + head * KD + c4 * 4);
        *(float4*)(q_lds + qrow * KD + c4 * 4) = v;
    }
    // ---- stage transposed position embeddings (zero-padded to N=32)
    for (int i = tid; i < 1024; i += 256) {
        int r = i >> 5, kd = i & 31;
        ehT[i] = (r < 23) ? emb_h[kd * 23 + r] : 0.f;
        ewT[i] = (r < 23) ? emb_w[kd * 23 + r] : 0.f;
    }
    // ---- gather halo'd key/value window, zero-fill out-of-bounds (== pad)
    if (tid < NPOS) {
        int p = tid, ky = p / KVK, kx = p % KVK;
        int gy = hy * 8 - HALO + ky, gx = wx * 8 - HALO + kx;
        bool ok = (gy >= 0) & (gy < HDIM) & (gx >= 0) & (gx < HDIM);
        const float* src = qkv + (((long)(b * HDIM + gy)) * HDIM + gx) * QKVN
                               + CC + head * 64;
        #pragma unroll
        for (int c = 0; c < KD; c += 4) {
            float4 kk = ok ? *(const float4*)(src + c)      : make_float4(0, 0, 0, 0);
            float4 vv = ok ? *(const float4*)(src + KD + c) : make_float4(0, 0, 0, 0);
            *(float4*)(k_lds + p * KD + c) = kk;
            vT_lds[(c + 0) * NPOS + p] = vv.x;   // V stored transposed -> B reads
            vT_lds[(c + 1) * NPOS + p] = vv.y;   // are contiguous float2 in K
            vT_lds[(c + 2) * NPOS + p] = vv.z;
            vT_lds[(c + 3) * NPOS + p] = vv.w;
        }
    }
    __syncthreads();

    // ---- relative logits via WMMA: REL = Q(64x32) @ emb(32x23, padded 32)
    {
        const float* embT = (wave < 4) ? ehT : ewT;
        float* rel = (wave < 4) ? relh : relw;
        int mt = wave & 3;
        v8f a0 = {}, a1 = {};
        #pragma unroll
        for (int step = 0; step < 8; ++step) {
            int k0 = 2 * lhalf + 4 * step;
            v2f a = *(const v2f*)(q_lds + (mt * 16 + lmod) * KD + k0);
            v2f b0 = *(const v2f*)(embT + lmod * 32 + k0);
            v2f b1 = *(const v2f*)(embT + (16 + lmod) * 32 + k0);
            a0 = wmma4(a, b0, a0);
            a1 = wmma4(a, b1, a1);
        }
        #pragma unroll
        for (int g = 0; g < 8; ++g) {
            int qrow = mt * 16 + g + 8 * lhalf;
            rel[qrow * 32 + lmod] = a0[g];
            rel[qrow * 32 + 16 + lmod] = a1[g];
        }
    }
    __syncthreads();

    // ---- scores: S = Q @ K^T (4 M-tiles x 9 N-tiles), + scale + rel logits
    const float scale = 0.1767766952966369f;     // 1/sqrt(32)
    for (int pr = wave; pr < 36; pr += 8) {      // wave-uniform
        int mt = pr & 3, nt = pr >> 2;
        v8f acc = {};
        #pragma unroll
        for (int step = 0; step < 8; ++step) {
            int k0 = 2 * lhalf + 4 * step;
            v2f a = *(const v2f*)(q_lds + (mt * 16 + lmod) * KD + k0);
            v2f bb = *(const v2f*)(k_lds + (nt * 16 + lmod) * KD + k0);
            acc = wmma4(a, bb, acc);
        }
        int col = nt * 16 + lmod;                // key position, < 144
        int ky = col / KVK, kx = col % KVK;
        #pragma unroll
        for (int g = 0; g < 8; ++g) {
            int qrow = mt * 16 + g + 8 * lhalf;
            int qy = qrow >> 3, qx = qrow & 7;
            // rel->abs map: abs[q][k] = rel[q][k - q + (kvk-1)]; index in [4,22]
            float logit = acc[g] * scale
                        + relh[qrow * 32 + (ky - qy + 11)]
                        + relw[qrow * 32 + (kx - qx + 11)];
            sc[qrow * NPOS + col] = logit;
        }
    }
    __syncthreads();

    // ---- softmax over 144 keys, one thread per query row
    if (tid < 64) {
        float* row = sc + tid * NPOS;
        float m = -1e30f;
        for (int j = 0; j < NPOS; ++j) m = fmaxf(m, row[j]);
        float s = 0.f;
        for (int j = 0; j < NPOS; ++j) { float e = __expf(row[j] - m); row[j] = e; s += e; }
        float inv = 1.f / s;
        for (int j = 0; j < NPOS; ++j) row[j] *= inv;
    }
    __syncthreads();

    // ---- out = P(64x144) @ V(144x32): wave -> (mt = wave&3, nt = wave>>2)
    {
        int mt = wave & 3, nt = wave >> 2;
        v8f acc = {};
        #pragma unroll
        for (int step = 0; step < 36; ++step) {
            int k0 = 2 * lhalf + 4 * step;
            v2f a = *(const v2f*)(sc + (mt * 16 + lmod) * NPOS + k0);
            v2f bb = *(const v2f*)(vT_lds + (nt * 16 + lmod) * NPOS + k0);
            acc = wmma4(a, bb, acc);
        }
        #pragma unroll
        for (int g = 0; g < 8; ++g) {
            int qrow = mt * 16 + g + 8 * lhalf;
            int qy = qrow >> 3, qx = qrow & 7;
            long grow = ((long)(b * HDIM + hy * 8 + qy)) * HDIM + wx * 8 + qx;
            attnout[grow * (long)CC + head * KD + nt * 16 + lmod] = acc[g];
        }
    }
}

// ---------------------------------------------------------------------------
extern "C" void kernel_launch(void* const* d_in, const int* in_sizes, int n_in,
                              void* d_out, int out_size, void* d_ws, size_t ws_size,
                              hipStream_t stream) {
    const float* x     = (const float*)d_in[0];   // [4,128,128,256]
    const float* w_q   = (const float*)d_in[1];   // [256,256]
    const float* w_kv  = (const float*)d_in[2];   // [256,512]
    const float* w_out = (const float*)d_in[3];   // [256,256]
    const float* pe_h  = (const float*)d_in[4];   // [32,23]
    const float* pe_w  = (const float*)d_in[5];   // [32,23]
    float* out = (float*)d_out;

    char* ws = (char*)d_ws;
    float* qkv     = (float*)ws;                                       // 65536x768
    float* attnout = (float*)(ws + (size_t)MROWS * QKVN * 4);          // 65536x256
    float* wcatT   = (float*)(ws + (size_t)MROWS * QKVN * 4
                                 + (size_t)MROWS * CC * 4);            // 768x256
    float* woutT   = wcatT + QKVN * CC;                                // 256x256

    prep_weights<<<QKVN, 256, 0, stream>>>(w_q, w_kv, w_out, wcatT, woutT);
    // fused Q + KV projection: [65536,256] x [256,768]
    gemm_wmma_f32<<<dim3(MROWS / 128, QKVN / 64), 256, 0, stream>>>(x, wcatT, qkv, QKVN);
    // attention per (b, head, 8x8 block): 4*8*16*16 = 8192 tiles
    halo_attn<<<8192, 256, 0, stream>>>(qkv, pe_h, pe_w, attnout);
    // output projection: [65536,256] x [256,256]
    gemm_wmma_f32<<<dim3(MROWS / 128, CC / 64), 256, 0, stream>>>(attnout, woutT, out, CC);
}